// SpatialPyramidPoolinglayer1d_57990648431310
// MI455X (gfx1250) — compile-verified
//
#include <hip/hip_runtime.h>
#include <cstdint>
#include <cstddef>

// Problem constants (match reference)
#define B_     32
#define T_     8192
#define C_     256
#define W_     21      // 1 + 4 + 16 windows
#define NSLOT  48      // chunk-blocks per batch: 16 (lvl0) + 4*4 (lvl1) + 16*1 (lvl2)
#define NWAVE  8       // 256 threads / wave32
#define STAGE_ROWS 4   // rows staged per wave per pipeline stage (8 async B128 issues)

// ---------------------------------------------------------------------------
// CDNA5 async global->LDS copy (VGLOBAL encoding, GV mode, tracked by ASYNCcnt)
// ---------------------------------------------------------------------------
__device__ __forceinline__ void async_ld_b128(unsigned lds_off, const float* g) {
  asm volatile("global_load_async_to_lds_b128 %0, %1, off"
               :: "v"(lds_off), "v"((unsigned long long)(uintptr_t)g)
               : "memory");
}
__device__ __forceinline__ void wait_async_le8() { asm volatile("s_wait_asynccnt 8" ::: "memory"); }
__device__ __forceinline__ void wait_async_0()   { asm volatile("s_wait_asynccnt 0" ::: "memory"); }
__device__ __forceinline__ void wait_ds_0()      { asm volatile("s_wait_dscnt 0"    ::: "memory"); }

// ---------------------------------------------------------------------------
// Kernel 1: per-(b,w) window params, bit-faithful to the JAX float32 math.
//   div_len = ceil(L/ndiv - 1e-8)           (f32)
//   start   = rint((L - div_len)/(ndiv-1) * div_index)   (rint == round-half-even)
// ---------------------------------------------------------------------------
__global__ void spp_params_kernel(const int* __restrict__ lengths,
                                  int* __restrict__ starts,
                                  int* __restrict__ lens) {
  const int idx = threadIdx.x;
  if (idx >= B_ * W_) return;
  const int b = idx / W_;
  const int w = idx % W_;
  const float L = (float)lengths[b];

  int ndiv, di;
  if (w == 0)      { ndiv = 1;  di = 0;      }
  else if (w < 5)  { ndiv = 4;  di = w - 1;  }
  else             { ndiv = 16; di = w - 5;  }

  const float dlf = ceilf(L / (float)ndiv - 1e-8f);
  const int   dl  = (int)dlf;
  int s = 0;
  if (ndiv > 1)
    s = (int)rintf((L - dlf) / (float)(ndiv - 1) * (float)di);

  starts[idx] = s;
  lens[idx]   = dl;
}

// ---------------------------------------------------------------------------
// Kernel 2: streaming max over a chunk of one window.
// Grid: B_*NSLOT blocks of 256 threads (8 wave32s).
// Each wave owns a private double-buffered LDS slice; lanes own 8 channels.
// Async global->LDS B128 loads double-buffered via ASYNCcnt.
// ---------------------------------------------------------------------------
__global__ __launch_bounds__(256) void spp_pool_kernel(const float* __restrict__ seq,
                                                       const int* __restrict__ starts,
                                                       const int* __restrict__ lens,
                                                       float* __restrict__ partial) {
  __shared__ float stage[NWAVE][2][STAGE_ROWS][C_]; // 64 KB staging
  __shared__ float red[NWAVE][C_];                  // 8 KB cross-wave combine

  const int tid  = threadIdx.x;
  const int wave = tid >> 5;
  const int lane = tid & 31;

  const int gid  = blockIdx.x;
  const int b    = gid / NSLOT;
  const int slot = gid - b * NSLOT;

  int w, chunk, nchunk;
  if (slot < 16)      { w = 0;                     chunk = slot;          nchunk = 16; }
  else if (slot < 32) { w = 1 + ((slot - 16) >> 2); chunk = (slot - 16) & 3; nchunk = 4; }
  else                { w = 5 + (slot - 32);       chunk = 0;             nchunk = 1;  }

  const int s  = starts[b * W_ + w];
  const int dl = lens  [b * W_ + w];
  const int q  = (dl + nchunk - 1) / nchunk;   // rows per chunk
  const int r0 = s + chunk * q;
  int r1 = s + dl;
  if (r0 + q < r1) r1 = r0 + q;
  if (r1 > T_)     r1 = T_;

  float acc[8];
#pragma unroll
  for (int e = 0; e < 8; ++e) acc[e] = -__builtin_huge_valf();

  // this wave's rows: r0 + wave + 8*i,  i = 0..n_i-1  (wave-uniform control flow)
  const int span = r1 - r0 - wave;
  const int n_i  = (span > 0) ? ((span + 7) >> 3) : 0;

  if (n_i > 0) {
    const int nstages = (n_i + STAGE_ROWS - 1) / STAGE_ROWS;
    const float* gbase = seq + (size_t)b * T_ * C_ + lane * 8;

    auto issue = [&](int S, int buf) {
#pragma unroll
      for (int j = 0; j < STAGE_ROWS; ++j) {
        int i = S * STAGE_ROWS + j;
        if (i > n_i - 1) i = n_i - 1;            // clamp: fixed 8 issues/stage
        const int r = r0 + wave + (i << 3);
        const float* g = gbase + (size_t)r * C_;
        const unsigned l =
            (unsigned)(unsigned long long)(uintptr_t)&stage[wave][buf][j][lane * 8];
        async_ld_b128(l,       g);
        async_ld_b128(l + 16u, g + 4);
      }
    };

    issue(0, 0);
    for (int S = 0; S < nstages; ++S) {
      const int buf = S & 1;
      if (S + 1 < nstages) {
        issue(S + 1, buf ^ 1);   // prefetch next stage into other buffer
        wait_async_le8();        // 8 newest remain -> current stage landed
      } else {
        wait_async_0();
      }
#pragma unroll
      for (int j = 0; j < STAGE_ROWS; ++j) {
        if (S * STAGE_ROWS + j < n_i) {
          const float4* p = (const float4*)&stage[wave][buf][j][lane * 8];
          const float4 x = p[0];
          const float4 y = p[1];
          acc[0] = fmaxf(acc[0], x.x); acc[1] = fmaxf(acc[1], x.y);
          acc[2] = fmaxf(acc[2], x.z); acc[3] = fmaxf(acc[3], x.w);
          acc[4] = fmaxf(acc[4], y.x); acc[5] = fmaxf(acc[5], y.y);
          acc[6] = fmaxf(acc[6], y.z); acc[7] = fmaxf(acc[7], y.w);
        }
      }
      wait_ds_0();  // LDS reads retired before this buffer is re-targeted (WAR)
    }
  }

  // cross-wave combine (one barrier), write this chunk's partial max
#pragma unroll
  for (int e = 0; e < 8; ++e) red[wave][lane * 8 + e] = acc[e];
  __syncthreads();

  const int c = tid;  // 0..255
  float m = red[0][c];
#pragma unroll
  for (int k = 1; k < NWAVE; ++k) m = fmaxf(m, red[k][c]);
  partial[((size_t)b * NSLOT + slot) * C_ + c] = m;
}

// ---------------------------------------------------------------------------
// Kernel 3: combine chunk partials per window, apply dl>0 zero rule.
// ---------------------------------------------------------------------------
__global__ __launch_bounds__(256) void spp_finalize_kernel(const float* __restrict__ partial,
                                                           const int* __restrict__ lens,
                                                           float* __restrict__ out) {
  const int idx = blockIdx.x * 256 + threadIdx.x;
  if (idx >= B_ * W_ * C_) return;
  const int c  = idx & (C_ - 1);
  const int bw = idx >> 8;        // C_ == 256
  const int w  = bw % W_;
  const int b  = bw / W_;

  int slot0, ns;
  if (w == 0)     { slot0 = 0;                ns = 16; }
  else if (w < 5) { slot0 = 16 + (w - 1) * 4; ns = 4;  }
  else            { slot0 = 32 + (w - 5);     ns = 1;  }

  float m = -__builtin_huge_valf();
  for (int k = 0; k < ns; ++k)
    m = fmaxf(m, partial[((size_t)b * NSLOT + slot0 + k) * C_ + c]);

  out[idx] = (lens[b * W_ + w] > 0) ? m : 0.0f;
}

// ---------------------------------------------------------------------------
extern "C" void kernel_launch(void* const* d_in, const int* in_sizes, int n_in,
                              void* d_out, int out_size, void* d_ws, size_t ws_size,
                              hipStream_t stream) {
  const float* seq     = (const float*)d_in[0];
  const int*   lengths = (const int*)d_in[1];
  float*       out     = (float*)d_out;

  // workspace layout: starts[B*W] | lens[B*W] | partial[B*NSLOT*C]
  int*   starts  = (int*)d_ws;
  int*   lens    = starts + B_ * W_;
  float* partial = (float*)(lens + B_ * W_);

  spp_params_kernel  <<<1,          B_ * W_, 0, stream>>>(lengths, starts, lens);
  spp_pool_kernel    <<<B_ * NSLOT, 256,     0, stream>>>(seq, starts, lens, partial);
  spp_finalize_kernel<<<B_ * W_,    256,     0, stream>>>(partial, lens, out);
}